// BalanceDropLoss_17987323036119
// MI455X (gfx1250) — compile-verified
//
#include <hip/hip_runtime.h>

// BalanceDropLoss for pred/target f32[524288, 40] -> scalar mean.
// Bandwidth-bound (~252 MB traffic ~= 10.8 us @ 23.3 TB/s).
// Pass 1 (column counts of target) is done as ones^T x target via
// V_WMMA_F32_16X16X4_F32 with exact f32 integer accumulation.

#define B_ROWS 524288
#define NCOLS  40
#define NELEM  (B_ROWS * NCOLS)        // 20,971,520
#define BALF   262144.0f               // 0.5 * B_ROWS

typedef __attribute__((ext_vector_type(2))) float v2f;
typedef __attribute__((ext_vector_type(8))) float v8f;

#define KA_BLOCKS  256
#define KA_THREADS 256
#define KB_BLOCKS  2048
#define KB_THREADS 256

// ---------------- Kernel 0: zero the 40 column counters ----------------
__global__ void k_init(unsigned* __restrict__ cnt) {
    if (threadIdx.x < NCOLS) cnt[threadIdx.x] = 0u;
}

// ---------------- Kernel A: pos_cnt[c] = sum_b target[b][c] via WMMA ----
// A = all-ones 16x4 (so D[m,n] = sum_k B[k,n] + C[m,n], independent of the
// hardware's K-slot assignment per lane). B tiles are 4 rows x 16 cols of
// target. The third tile (cols 32..47) is handled branchlessly: loads are
// clamped to column 39 (same cacheline as valid data) and lanes with
// n >= 8 are zeroed by a precomputed 0/1 mask -> no exec divergence.
__global__ void k_colsum(const float* __restrict__ tgt, unsigned* __restrict__ cnt) {
    const int lane   = threadIdx.x & 31;
    const int wave   = (blockIdx.x * blockDim.x + threadIdx.x) >> 5;
    const int nwaves = (gridDim.x * blockDim.x) >> 5;
    const int n      = lane & 15;   // N position (column within 16-wide tile)
    const int half   = lane >> 4;   // which half of the wave

    const int   c2   = 32 + ((n < 8) ? n : 7);          // clamped tail column
    const float msk2 = (n < 8) ? 1.0f : 0.0f;           // zero the pad lanes

    v2f a; a[0] = 1.0f; a[1] = 1.0f;      // all-ones A matrix
    v8f acc0 = {}; v8f acc1 = {}; v8f acc2 = {};

    const int CHUNKS = B_ROWS / 16;       // 16 rows per wave-iteration
    for (int ch = wave; ch < CHUNKS; ch += nwaves) {
        const int r0 = ch * 16;
#pragma unroll
        for (int u = 0; u < 4; ++u) {
            const float* rowA = tgt + (size_t)(r0 + 4 * u + half)     * NCOLS; // k slots 0,1
            const float* rowB = tgt + (size_t)(r0 + 4 * u + 2 + half) * NCOLS; // k slots 2,3
            {   // cols 0..15
                v2f b; b[0] = rowA[n]; b[1] = rowB[n];
                acc0 = __builtin_amdgcn_wmma_f32_16x16x4_f32(
                           false, a, false, b, (short)0, acc0, false, false);
            }
            {   // cols 16..31
                v2f b; b[0] = rowA[16 + n]; b[1] = rowB[16 + n];
                acc1 = __builtin_amdgcn_wmma_f32_16x16x4_f32(
                           false, a, false, b, (short)0, acc1, false, false);
            }
            {   // cols 32..39 (+8 zero-masked), branchless
                v2f b; b[0] = rowA[c2] * msk2; b[1] = rowB[c2] * msk2;
                acc2 = __builtin_amdgcn_wmma_f32_16x16x4_f32(
                           false, a, false, b, (short)0, acc2, false, false);
            }
        }
    }
    // Every output row M holds the same column sums; element 0, lanes 0..15
    // hold (M=0, N=lane). Counts are exact integers in f32 -> commit with
    // deterministic integer atomics.
    if (lane < 16) {
        atomicAdd(&cnt[lane],      (unsigned)(acc0[0] + 0.5f));
        atomicAdd(&cnt[16 + lane], (unsigned)(acc1[0] + 0.5f));
        if (lane < 8)
            atomicAdd(&cnt[32 + lane], (unsigned)(acc2[0] + 0.5f));
    }
}

// ---------------- Kernel B: weighted BCE + per-block partial sums -------
__global__ void k_loss(const float* __restrict__ pred, const float* __restrict__ tgt,
                       const unsigned* __restrict__ cnt, float* __restrict__ partial) {
    __shared__ float s_pg[NCOLS], s_ng[NCOLS], s_mw[NCOLS], s_nw[NCOLS];
    __shared__ float s_red[KB_THREADS];

    if (threadIdx.x < NCOLS) {
        const float ps = (float)cnt[threadIdx.x];
        const float ns = (float)B_ROWS - ps;
        const int pg = (ps >= BALF) ? 1 : 0;
        const int ng = (ns >  BALF) ? 1 : 0;
        const float nmaj = pg ? ps : ns;
        const float nmin = ng ? ps : ns;
        s_pg[threadIdx.x] = (float)pg;
        s_ng[threadIdx.x] = (float)ng;
        s_mw[threadIdx.x] = BALF / fmaxf(nmaj, 1.0f);
        // fold the (n_min > 0) condition into the scale (1.0 == no-op)
        s_nw[threadIdx.x] = (nmin > 0.0f)
                          ? (((float)B_ROWS - BALF) / fmaxf(nmin, 1.0f)) : 1.0f;
    }
    __syncthreads();

    const int N4 = NELEM / 4;          // row length 40 % 4 == 0: no row-crossing
    const float4* p4 = (const float4*)pred;
    const float4* t4 = (const float4*)tgt;

    float sum = 0.0f;
    for (int i = blockIdx.x * blockDim.x + threadIdx.x; i < N4;
         i += gridDim.x * blockDim.x) {
        const float4 x = p4[i];
        const float4 t = t4[i];
        const int c0 = (i % 10) * 4;   // (4*i) % 40
        const float xv[4] = {x.x, x.y, x.z, x.w};
        const float tv[4] = {t.x, t.y, t.z, t.w};
#pragma unroll
        for (int j = 0; j < 4; ++j) {
            const int   c  = c0 + j;
            const float xx = xv[j];
            const float tt = tv[j];
            const float sig = 1.0f / (1.0f + expf(-xx));
            const float g   = fabsf(sig - tt);
            const bool maj  = (tt == s_pg[c]);   // values are exact 0/1 floats
            const bool minr = (tt == s_ng[c]);
            float w = 1.0f;
            if (maj)  w = (g < 0.1f) ? 0.0f : s_mw[c];  // drop easy majority + rescale
            if (minr) w *= s_nw[c];
            const float bce = fmaxf(xx, 0.0f) - xx * tt + log1pf(expf(-fabsf(xx)));
            sum += bce * w;
        }
    }

    s_red[threadIdx.x] = sum;
    __syncthreads();
    for (int off = KB_THREADS / 2; off > 0; off >>= 1) {
        if (threadIdx.x < off) s_red[threadIdx.x] += s_red[threadIdx.x + off];
        __syncthreads();
    }
    if (threadIdx.x == 0) partial[blockIdx.x] = s_red[0];
}

// ---------------- Kernel C: deterministic finalize ----------------------
__global__ void k_final(const float* __restrict__ partial, int n, float* __restrict__ out) {
    __shared__ double s[256];
    double acc = 0.0;
    for (int i = threadIdx.x; i < n; i += 256) acc += (double)partial[i];
    s[threadIdx.x] = acc;
    __syncthreads();
    for (int off = 128; off > 0; off >>= 1) {
        if (threadIdx.x < off) s[threadIdx.x] += s[threadIdx.x + off];
        __syncthreads();
    }
    if (threadIdx.x == 0) out[0] = (float)(s[0] / (double)NELEM);
}

extern "C" void kernel_launch(void* const* d_in, const int* in_sizes, int n_in,
                              void* d_out, int out_size, void* d_ws, size_t ws_size,
                              hipStream_t stream) {
    const float* pred = (const float*)d_in[0];
    const float* tgt  = (const float*)d_in[1];
    unsigned* cnt     = (unsigned*)d_ws;                  // 40 x u32 counters
    float*    partial = (float*)((char*)d_ws + 256);      // KB_BLOCKS x f32
    float*    out     = (float*)d_out;

    k_init  <<<1,          64,         0, stream>>>(cnt);
    k_colsum<<<KA_BLOCKS,  KA_THREADS, 0, stream>>>(tgt, cnt);
    k_loss  <<<KB_BLOCKS,  KB_THREADS, 0, stream>>>(pred, tgt, cnt, partial);
    k_final <<<1,          256,        0, stream>>>(partial, KB_BLOCKS, out);
}